// mLSTMBlock_16965120819374
// MI455X (gfx1250) — compile-verified
//
#include <hip/hip_runtime.h>
#include <hip/hip_bf16.h>
#include <math.h>

#define B_  64
#define I_  1024
#define HID 2048
#define H_  8
#define D_  256
#define KS_ 4
#define EPS 1e-5f

typedef float v2f __attribute__((ext_vector_type(2)));
typedef float v8f __attribute__((ext_vector_type(8)));

// ---------------------------------------------------------------------------
// Block-wide sum reduction through LDS (n = blockDim, power of two)
// ---------------------------------------------------------------------------
__device__ __forceinline__ float block_sum(float v, float* red, int t, int n) {
    red[t] = v; __syncthreads();
    for (int o = n >> 1; o > 0; o >>= 1) {
        if (t < o) red[t] += red[t + o];
        __syncthreads();
    }
    float r = red[0]; __syncthreads();
    return r;
}

__device__ __forceinline__ float sigmoidf_(float v) { return 1.f / (1.f + __expf(-v)); }
__device__ __forceinline__ float siluf_(float v)    { return v * sigmoidf_(v); }

// ---------------------------------------------------------------------------
// LayerNorm: x[64,1024] -> xn[64,1024]
// ---------------------------------------------------------------------------
__global__ __launch_bounds__(256) void ln_kernel(const float* __restrict__ x,
                                                 const float* __restrict__ w,
                                                 const float* __restrict__ b,
                                                 float* __restrict__ xn) {
    __shared__ float red[256];
    const int bi = blockIdx.x, t = threadIdx.x;
    const float* xr = x + bi * I_;
    float s = 0.f;
    for (int i = t; i < I_; i += 256) s += xr[i];
    const float mu = block_sum(s, red, t, 256) * (1.f / I_);
    float s2 = 0.f;
    for (int i = t; i < I_; i += 256) { float d = xr[i] - mu; s2 += d * d; }
    const float var = block_sum(s2, red, t, 256) * (1.f / I_);
    const float rs = rsqrtf(var + EPS);
    for (int i = t; i < I_; i += 256)
        xn[bi * I_ + i] = (xr[i] - mu) * rs * w[i] + b[i];
}

// ---------------------------------------------------------------------------
// Generic fp32 WMMA GEMM:  C[64, cN*grid.y] = act((A @ W + bias) * scale) [+R]
//   A element (m,k)  at  A + batch*aBatch + m*lda + k
//   W element (k,n)  at  W + batch*wBatch + k*wK + n*wN        (n local)
//   C element (m,n)  at  C + m*ldc + nGlobal,  nGlobal = batch*cN + n
// One wave (32 threads) per 16-wide N strip; 4 accumulators cover M=64.
// Uses V_WMMA_F32_16X16X4_F32 (ISA 7.12.2 layouts).
// ---------------------------------------------------------------------------
template <int ACT, bool RESID>   // ACT: 0 none, 1 silu, 2 sigmoid
__global__ __launch_bounds__(32) void gemm_wmma(
        const float* __restrict__ A, int lda, int aBatch,
        const float* __restrict__ W, long wK, long wN, long wBatch,
        const float* __restrict__ bias,
        const float* __restrict__ R, int ldr,
        float* __restrict__ C, int ldc,
        int K, int cN, float scale) {
    const int lane = threadIdx.x;
    const int half = lane >> 4;          // selects K +0/+2 (A) and rows K,K+2 (B)
    const int lr   = lane & 15;          // M row within tile / N column within strip
    const int batch = blockIdx.y;
    const int nLocal  = blockIdx.x * 16 + lr;
    const int nGlobal = batch * cN + nLocal;

    const float* Ab = A + (long)batch * aBatch;
    const float* Wb = W + (long)batch * wBatch;

    v8f acc[4];
#pragma unroll
    for (int i = 0; i < 4; ++i) acc[i] = 0.f;

    for (int k = 0; k < K; k += 4) {
        const int kk = k + 2 * half;
        // B 4x16: row kk in lanes 0-15 / row kk (=k+2) in lanes 16-31, rows kk+1/kk+3 in VGPR1
        v2f bv;
        bv.x = Wb[(long)kk * wK + (long)nLocal * wN];
        bv.y = Wb[(long)(kk + 1) * wK + (long)nLocal * wN];
        __builtin_prefetch(&Wb[(long)(kk + 64) * wK + (long)nLocal * wN], 0, 1);
#pragma unroll
        for (int mt = 0; mt < 4; ++mt) {
            const int row = mt * 16 + lr;
            const float* ap = Ab + (long)row * lda + kk;   // contiguous pair
            v2f av; av.x = ap[0]; av.y = ap[1];
            acc[mt] = __builtin_amdgcn_wmma_f32_16x16x4_f32(
                    false, av, false, bv, (short)0, acc[mt], false, false);
        }
    }

    const float bval = bias ? bias[nGlobal] : 0.f;
#pragma unroll
    for (int mt = 0; mt < 4; ++mt) {
#pragma unroll
        for (int r = 0; r < 8; ++r) {
            const int row = mt * 16 + half * 8 + r;        // C/D: VGPR r -> M=r (+8 for hi half)
            float v = (acc[mt][r] + bval) * scale;
            if (ACT == 1) v = siluf_(v);
            if (ACT == 2) v = sigmoidf_(v);
            if (RESID) v += R[(long)row * ldr + nGlobal];
            C[(long)row * ldc + nGlobal] = v;
        }
    }
}

// ---------------------------------------------------------------------------
// Input/forget gate scalars + stabilized exponential gating, per (b,h)
// ---------------------------------------------------------------------------
__global__ __launch_bounds__(128) void gates_kernel(
        const float* __restrict__ xc,
        const float* __restrict__ W_i, const float* __restrict__ b_i,
        const float* __restrict__ W_f, const float* __restrict__ b_f,
        const float* __restrict__ m_prev,
        float* __restrict__ m_out, float* __restrict__ ig, float* __restrict__ fg) {
    __shared__ float red[128];
    const int bh = blockIdx.x, b = bh >> 3, h = bh & 7, t = threadIdx.x;
    const float* xr = xc + b * HID;
    float si = 0.f, sf = 0.f;
    for (int i = t; i < HID; i += 128) {
        const float xv = xr[i];
        si += xv * W_i[i * H_ + h];
        sf += xv * W_f[i * H_ + h];
    }
    const float it = block_sum(si, red, t, 128) + b_i[h];
    const float ft = block_sum(sf, red, t, 128) + b_f[h];
    if (t == 0) {
        const float mp = m_prev[bh];
        const float mt = fmaxf(ft + mp, it);
        m_out[bh] = mt;
        ig[bh] = __expf(it - mt);
        fg[bh] = __expf(ft + mp - mt);
    }
}

// ---------------------------------------------------------------------------
// Fused state update per (b,h): n_t, den, c_t stream, num, h_t, GroupNorm.
// One pass over the 256x256 cell state (the HBM-bound part of this op).
// q/k head vectors are staged into LDS with gfx1250 async global->LDS copies
// (GLOBAL_LOAD_ASYNC_TO_LDS_B32, tracked by ASYNCcnt).
// ---------------------------------------------------------------------------
__global__ __launch_bounds__(256) void state_kernel(
        const float* __restrict__ c_prev, const float* __restrict__ n_prev,
        const float* __restrict__ qws, const float* __restrict__ kws,
        const float* __restrict__ vws, const float* __restrict__ ows,
        const float* __restrict__ ig, const float* __restrict__ fg,
        const float* __restrict__ gn_w, const float* __restrict__ gn_b,
        float* __restrict__ c_out, float* __restrict__ n_out,
        float* __restrict__ h_out, float* __restrict__ gn_out) {
    __shared__ float sq[D_], sk[D_], red[256];
    const int bh = blockIdx.x, b = bh >> 3, h = bh & 7, t = threadIdx.x;
    const float gi = ig[bh], gf = fg[bh];
    const int col = h * D_ + t;

    // Async memory->LDS staging of q and k rows (ASYNCcnt path). The generic
    // pointer's low 32 bits are the wave-relative LDS byte offset (flat LDS
    // aperture truncation rule).
    {
        unsigned int lq = (unsigned int)(unsigned long long)(&sq[t]);
        unsigned int lk = (unsigned int)(unsigned long long)(&sk[t]);
        unsigned long long gq = (unsigned long long)(&qws[b * HID + col]);
        unsigned long long gk = (unsigned long long)(&kws[b * HID + col]);
        asm volatile("global_load_async_to_lds_b32 %0, %1, off"
                     :: "v"(lq), "v"(gq) : "memory");
        asm volatile("global_load_async_to_lds_b32 %0, %1, off"
                     :: "v"(lk), "v"(gk) : "memory");
        asm volatile("s_wait_asynccnt 0x0" ::: "memory");
    }
    __syncthreads();

    const float qv = sq[t];
    const float kv = sk[t];
    const float vv = vws[b * HID + col];

    // n_t and denominator
    const float nv = gf * n_prev[(long)bh * D_ + t] + gi * kv;
    n_out[(long)bh * D_ + t] = nv;
    const float den = fmaxf(block_sum(nv * qv, red, t, 256), 1.f);

    // c_t row d = t, fused readout num[d] = sum_k c_t[d,k] * q[k]
    const float4* cp = (const float4*)(c_prev + ((long)bh * D_ + t) * D_);
    float4*       co = (float4*)(c_out + ((long)bh * D_ + t) * D_);
    const float iv = gi * vv;
    float num = 0.f;
#pragma unroll 4
    for (int kk = 0; kk < D_ / 4; ++kk) {
        const float4 c4 = cp[kk];
        const int k0 = kk * 4;
        float4 o4;
        o4.x = gf * c4.x + iv * sk[k0 + 0];
        o4.y = gf * c4.y + iv * sk[k0 + 1];
        o4.z = gf * c4.z + iv * sk[k0 + 2];
        o4.w = gf * c4.w + iv * sk[k0 + 3];
        num += o4.x * sq[k0 + 0] + o4.y * sq[k0 + 1] + o4.z * sq[k0 + 2] + o4.w * sq[k0 + 3];
        co[kk] = o4;
    }

    const float hv = ows[b * HID + col] * num / den;
    h_out[b * HID + col] = hv;

    // GroupNorm over the head (256 values, one per thread)
    const float mu = block_sum(hv, red, t, 256) * (1.f / D_);
    const float dh = hv - mu;
    const float var = block_sum(dh * dh, red, t, 256) * (1.f / D_);
    gn_out[b * HID + col] = dh * rsqrtf(var + EPS) * gn_w[col] + gn_b[col];
}

// ---------------------------------------------------------------------------
// u = (gn + x_skip) * silu(xr)  (elementwise over [64,2048])
// ---------------------------------------------------------------------------
__global__ __launch_bounds__(256) void combine_kernel(
        const float* __restrict__ gn, const float* __restrict__ skip,
        const float* __restrict__ xr, float* __restrict__ u) {
    const int i = blockIdx.x * 256 + threadIdx.x;
    u[i] = (gn[i] + skip[i]) * siluf_(xr[i]);
}

// ---------------------------------------------------------------------------
extern "C" void kernel_launch(void* const* d_in, const int* in_sizes, int n_in,
                              void* d_out, int out_size, void* d_ws, size_t ws_size,
                              hipStream_t stream) {
    const float* x      = (const float*)d_in[0];
    const float* c_prev = (const float*)d_in[2];
    const float* n_prev = (const float*)d_in[3];
    const float* m_prev = (const float*)d_in[4];
    const float* ln_w   = (const float*)d_in[5];
    const float* ln_b   = (const float*)d_in[6];
    const float* W_ul   = (const float*)d_in[7];
    const float* b_ul   = (const float*)d_in[8];
    const float* W_ur   = (const float*)d_in[9];
    const float* b_ur   = (const float*)d_in[10];
    const float* W_conv = (const float*)d_in[11];
    const float* b_conv = (const float*)d_in[12];
    const float* W_skip = (const float*)d_in[13];
    const float* b_skip = (const float*)d_in[14];
    const float* W_q    = (const float*)d_in[15];
    const float* b_q    = (const float*)d_in[16];
    const float* W_k    = (const float*)d_in[17];
    const float* b_k    = (const float*)d_in[18];
    const float* W_v    = (const float*)d_in[19];
    const float* b_v    = (const float*)d_in[20];
    const float* W_i    = (const float*)d_in[21];
    const float* b_i    = (const float*)d_in[22];
    const float* W_f    = (const float*)d_in[23];
    const float* b_f    = (const float*)d_in[24];
    const float* W_o    = (const float*)d_in[25];
    const float* b_o    = (const float*)d_in[26];
    const float* W_down = (const float*)d_in[27];
    const float* b_down = (const float*)d_in[28];
    const float* gn_w   = (const float*)d_in[29];
    const float* gn_b   = (const float*)d_in[30];

    // Output layout (tuple concat): final, h_t, c_t, n_t, m_t
    float* out       = (float*)d_out;
    float* out_final = out;
    float* out_h     = out_final + B_ * I_;
    float* out_c     = out_h + B_ * HID;
    float* out_n     = out_c + (long)B_ * H_ * D_ * D_;
    float* out_m     = out_n + B_ * HID;

    // Workspace carve-up (floats)
    float* ws  = (float*)d_ws;
    float* xn  = ws;                 // 64*1024
    float* xl  = xn  + B_ * I_;      // 64*2048 each below
    float* xr  = xl  + B_ * HID;
    float* xc  = xr  + B_ * HID;
    float* skp = xc  + B_ * HID;
    float* ogt = skp + B_ * HID;
    float* qws = ogt + B_ * HID;
    float* kws = qws + B_ * HID;
    float* vws = kws + B_ * HID;
    float* gnw = vws + B_ * HID;
    float* uws = gnw + B_ * HID;
    float* igs = uws + B_ * HID;     // 512
    float* fgs = igs + B_ * H_;      // 512

    const dim3 wv(32);

    // 1) LayerNorm
    ln_kernel<<<B_, 256, 0, stream>>>(x, ln_w, ln_b, xn);

    // 2,3) Up projections  xl = xn@W_ul+b,  xr = xn@W_ur+b       [64,1024]x[1024,2048]
    gemm_wmma<0, false><<<dim3(HID / 16, 1), wv, 0, stream>>>(
        xn, I_, 0, W_ul, HID, 1, 0, b_ul, nullptr, 0, xl, HID, I_, HID, 1.f);
    gemm_wmma<0, false><<<dim3(HID / 16, 1), wv, 0, stream>>>(
        xn, I_, 0, W_ur, HID, 1, 0, b_ur, nullptr, 0, xr, HID, I_, HID, 1.f);

    // 4) Causal conv (len-1 seq => last tap only): xc = silu(xl @ W_conv[:,:,3].T + b_conv)
    //    W element (k=i, n=o) at W_conv + 3 + i*KS + o*HID*KS
    gemm_wmma<1, false><<<dim3(HID / 16, 1), wv, 0, stream>>>(
        xl, HID, 0, W_conv + (KS_ - 1), KS_, (long)HID * KS_, 0, b_conv,
        nullptr, 0, xc, HID, HID, HID, 1.f);

    // 5) skip = xc @ W_skip + b
    gemm_wmma<0, false><<<dim3(HID / 16, 1), wv, 0, stream>>>(
        xc, HID, 0, W_skip, HID, 1, 0, b_skip, nullptr, 0, skp, HID, HID, HID, 1.f);

    // 6) o = sigmoid(xl @ W_o + b)
    gemm_wmma<2, false><<<dim3(HID / 16, 1), wv, 0, stream>>>(
        xl, HID, 0, W_o, HID, 1, 0, b_o, nullptr, 0, ogt, HID, HID, HID, 1.f);

    // 7-9) Block-diagonal q/k/v: per-head [64,256]x[256,256], grid.y = heads
    gemm_wmma<0, false><<<dim3(D_ / 16, H_), wv, 0, stream>>>(
        xc, HID, D_, W_q, D_, 1, (long)D_ * D_, b_q, nullptr, 0, qws, HID, D_, D_, 1.f);
    gemm_wmma<0, false><<<dim3(D_ / 16, H_), wv, 0, stream>>>(
        xc, HID, D_, W_k, D_, 1, (long)D_ * D_, b_k, nullptr, 0, kws, HID, D_, D_, 0.0625f);
    gemm_wmma<0, false><<<dim3(D_ / 16, H_), wv, 0, stream>>>(
        xl, HID, D_, W_v, D_, 1, (long)D_ * D_, b_v, nullptr, 0, vws, HID, D_, D_, 1.f);

    // 10) gates + stabilized exponentials (+ m_t output)
    gates_kernel<<<B_ * H_, 128, 0, stream>>>(xc, W_i, b_i, W_f, b_f, m_prev,
                                              out_m, igs, fgs);

    // 11) fused state update / readout / GroupNorm (HBM-bound c stream)
    state_kernel<<<B_ * H_, 256, 0, stream>>>(c_prev, n_prev, qws, kws, vws, ogt,
                                              igs, fgs, gn_w, gn_b,
                                              out_c, out_n, out_h, gnw);

    // 12) u = (gn + skip) * silu(xr)
    combine_kernel<<<(B_ * HID) / 256, 256, 0, stream>>>(gnw, skp, xr, uws);

    // 13) final = u @ W_down + b_down + x
    gemm_wmma<0, true><<<dim3(I_ / 16, 1), wv, 0, stream>>>(
        uws, HID, 0, W_down, I_, 1, 0, b_down, x, I_, out_final, I_, HID, I_, 1.f);
}